// SimpleSplitFF_32418413150502
// MI455X (gfx1250) — compile-verified
//
#include <hip/hip_runtime.h>
#include <math.h>

// ---- problem constants (match reference) ----
#define DM   1024      // model dim
#define NE   32        // experts
#define FE   128       // expert hidden
#define TT   4         // slots per group
#define BSZ  4         // batch
#define SEQ  4096      // sequence
#define NG   (SEQ/TT)  // groups per batch = 1024
#define NROWS (BSZ*SEQ) // 16384 token rows
#define NTASK (BSZ*TT*NE) // 512 selected (b,t,e) tasks

typedef __attribute__((ext_vector_type(2))) float v2f;
typedef __attribute__((ext_vector_type(8))) float v8f;

// ---------------------------------------------------------------------------
// Kernel 1: logits = x(16384x1024) @ controller(1024x32), fp32 WMMA 16x16x4.
// One wave per 16-row tile; it owns BOTH 16-wide expert tiles (2 accumulators,
// shared A fragment) so every x element is streamed from HBM exactly once.
// x loads are non-temporal: zero reuse, keep L2 for controller/f1/f2/logits.
// A-frag (16x4 f32): lanes 0-15 -> K0/K1 in v0/v1, lanes 16-31 -> K2/K3.
// B-frag (4x16 f32): v0 = row K0 (lo lanes) / K2 (hi lanes), v1 = K1 / K3.
// C/D (16x16 f32): VGPR i -> row i (lanes 0-15), row i+8 (lanes 16-31).
// ---------------------------------------------------------------------------
__global__ void logits_wmma_kernel(const float* __restrict__ x,
                                   const float* __restrict__ controller,
                                   float* __restrict__ logits)
{
    const int waveId = blockIdx.x * (blockDim.x >> 5) + (threadIdx.x >> 5);
    const int row0 = waveId << 4;           // 1024 row tiles of 16

    const int lane = threadIdx.x & 31;
    const int half = lane >> 4;             // 0: K pair (k,k+1), 1: (k+2,k+3)
    const int ml   = lane & 15;             // row (A) / col (B,D)

    const float* __restrict__ xrow = x + (size_t)(row0 + ml) * DM;

    v8f acc0 = {};                          // experts 0..15
    v8f acc1 = {};                          // experts 16..31
    for (int k = 0; k < DM; k += 4) {
        const int ka = k + 2 * half;
        // A fragment: 8B-aligned pair, streamed once -> non-temporal
        v2f a = __builtin_nontemporal_load((const v2f*)(xrow + ka));

        const float* __restrict__ crow = controller + (size_t)ka * NE + ml;
        v2f b0, b1;
        b0.x = crow[0];        // K=ka,   N=ml
        b0.y = crow[NE];       // K=ka+1, N=ml
        b1.x = crow[16];       // K=ka,   N=16+ml
        b1.y = crow[NE + 16];  // K=ka+1, N=16+ml

        acc0 = __builtin_amdgcn_wmma_f32_16x16x4_f32(
                   false, a, false, b0, (short)0, acc0, false, false);
        acc1 = __builtin_amdgcn_wmma_f32_16x16x4_f32(
                   false, a, false, b1, (short)0, acc1, false, false);
    }

#pragma unroll
    for (int i = 0; i < 8; ++i) {
        const int m = i + 8 * half;
        logits[(size_t)(row0 + m) * NE + ml]      = acc0[i];
        logits[(size_t)(row0 + m) * NE + 16 + ml] = acc1[i];
    }
}

// ---------------------------------------------------------------------------
// Kernel 2: per (b,t,e) task, reduce over g=1024 groups:
//   g* = argmax_g logit(b,g,t,e);  p = 1 / sum_g exp(logit - max)
// (tie-break linspace is constant over g -> cancels in softmax & argmax)
// One 256-thread block per task.
// ---------------------------------------------------------------------------
__global__ void select_kernel(const float* __restrict__ logits,
                              int* __restrict__ sel_g,
                              float* __restrict__ sel_p)
{
    const int tau = blockIdx.x;
    const int b = tau / (TT * NE);
    const int r = tau % (TT * NE);
    const int t = r / NE;
    const int e = r % NE;
    const int tid = threadIdx.x;

    __shared__ float smax[256];
    __shared__ int   sarg[256];
    __shared__ float ssum[256];

    float lmax = -3.4e38f;
    int   larg = 0;
    for (int g = tid; g < NG; g += 256) {
        const float l = logits[(size_t)(b * SEQ + g * TT + t) * NE + e];
        if (l > lmax) { lmax = l; larg = g; }
    }
    smax[tid] = lmax; sarg[tid] = larg;
    __syncthreads();
    for (int s = 128; s > 0; s >>= 1) {
        if (tid < s) {
            const float o = smax[tid + s];
            if (o > smax[tid] || (o == smax[tid] && sarg[tid + s] < sarg[tid])) {
                smax[tid] = o; sarg[tid] = sarg[tid + s];
            }
        }
        __syncthreads();
    }
    const float gmax = smax[0];

    float lsum = 0.f;
    for (int g = tid; g < NG; g += 256) {
        const float l = logits[(size_t)(b * SEQ + g * TT + t) * NE + e];
        lsum += expf(l - gmax);
    }
    ssum[tid] = lsum;
    __syncthreads();
    for (int s = 128; s > 0; s >>= 1) {
        if (tid < s) ssum[tid] += ssum[tid + s];
        __syncthreads();
    }
    if (tid == 0) {
        sel_g[tau] = sarg[0];
        sel_p[tau] = 1.0f / ssum[0];   // exp(max-max)/sum
    }
}

// ---------------------------------------------------------------------------
// Kernel 3: one 128-thread block per (b,t,e) task.
//   u[d]  = sum_{t': sel(b,t',e)==g} p_{t'} * x[b, g*4+t', d]   (merge pre-ReLU)
//   h[f]  = relu( u . f1[:,e,f] + bias[e,f] )
//   out[b, g*4+t, :] += p_t * (h . f2[e,:,:])     (atomic scatter)
// ---------------------------------------------------------------------------
__global__ void expert_kernel(const float* __restrict__ x,
                              const float* __restrict__ f1,
                              const float* __restrict__ bias,
                              const float* __restrict__ f2,
                              const int* __restrict__ sel_g,
                              const float* __restrict__ sel_p,
                              float* __restrict__ out)
{
    const int tau = blockIdx.x;
    const int b = tau / (TT * NE);
    const int r = tau % (TT * NE);
    const int t = r / NE;
    const int e = r % NE;
    const int tid = threadIdx.x;          // 0..127

    const int   g = sel_g[tau];
    const float p = sel_p[tau];

    // selections of sibling slots for the same (b, e)
    int   gw[TT];
    float pw[TT];
#pragma unroll
    for (int ts = 0; ts < TT; ++ts) {
        const int sib = b * (TT * NE) + ts * NE + e;
        gw[ts] = sel_g[sib];
        pw[ts] = sel_p[sib];
    }

    __shared__ float u[DM];
    __shared__ float hs[FE];

    for (int d = tid; d < DM; d += 128) {
        float s = 0.f;
#pragma unroll
        for (int ts = 0; ts < TT; ++ts) {
            if (gw[ts] == g) {
                s += pw[ts] * x[(size_t)(b * SEQ + g * TT + ts) * DM + d];
            }
        }
        u[d] = s;
    }
    __syncthreads();

    // up-projection: thread tid owns hidden unit f = tid (f1 reads coalesced)
    float h = 0.f;
    for (int d = 0; d < DM; ++d) {
        h = fmaf(u[d], f1[((size_t)d * NE + e) * FE + tid], h);
    }
    h = fmaxf(h + bias[(size_t)e * FE + tid], 0.f);
    hs[tid] = h;
    __syncthreads();

    // down-projection + scatter (f2 reads coalesced over d)
    const float* __restrict__ f2e = f2 + (size_t)e * FE * DM;
    float* __restrict__ orow = out + (size_t)(b * SEQ + g * TT + t) * DM;
    for (int d = tid; d < DM; d += 128) {
        float acc = 0.f;
#pragma unroll 8
        for (int f = 0; f < FE; ++f) {
            acc = fmaf(hs[f], f2e[(size_t)f * DM + d], acc);
        }
        atomicAdd(&orow[d], p * acc);
    }
}

// ---------------------------------------------------------------------------
extern "C" void kernel_launch(void* const* d_in, const int* in_sizes, int n_in,
                              void* d_out, int out_size, void* d_ws, size_t ws_size,
                              hipStream_t stream)
{
    const float* x          = (const float*)d_in[0];   // (4,4096,1024)
    const float* controller = (const float*)d_in[1];   // (1024,32)
    const float* f1         = (const float*)d_in[2];   // (1024,32,128)
    const float* bias       = (const float*)d_in[3];   // (32,128)
    const float* f2         = (const float*)d_in[4];   // (32,128,1024)
    float* out = (float*)d_out;                        // (4,4096,1024)

    // workspace layout
    float* ws_logits = (float*)d_ws;                           // NROWS*NE floats (2 MB)
    int*   ws_selg   = (int*)((char*)d_ws + (size_t)NROWS * NE * sizeof(float));
    float* ws_selp   = (float*)((char*)ws_selg + (size_t)NTASK * sizeof(int));

    // zero output (sparse scatter target)
    hipMemsetAsync(out, 0, (size_t)out_size * sizeof(float), stream);

    // 1) logits GEMM: 1024 row tiles, each wave owns both expert tiles
    logits_wmma_kernel<<<(NROWS / 16) / 8, 256, 0, stream>>>(
        x, controller, ws_logits);

    // 2) argmax + softmax denominator over groups, per (b,t,e)
    select_kernel<<<NTASK, 256, 0, stream>>>(ws_logits, ws_selg, ws_selp);

    // 3) sparse expert apply + scatter
    expert_kernel<<<NTASK, 128, 0, stream>>>(x, f1, bias, f2, ws_selg, ws_selp, out);
}